// AttentionRefinementModule_25838523253283
// MI455X (gfx1250) — compile-verified
//
#include <hip/hip_runtime.h>
#include <hip/hip_bf16.h>
#include <stdint.h>

typedef __attribute__((ext_vector_type(2))) float v2f;
typedef __attribute__((ext_vector_type(8))) float v8f;

#define NHEAD 8
#define C_IN 16      // 2*NHEAD
#define DC 32
#define T 256
#define B 8
#define H 16
#define W 32
#define L 512        // H*W
#define KTOT 400     // C_IN * 5 * 5
#define EPS 1e-5f

// ---------------------------------------------------------------------------
// Kernel 0: zero BN accumulators, compute unmasked count (ws is NOT re-poisoned
// by the harness between calls, so we must re-init every launch).
// ---------------------------------------------------------------------------
__global__ __launch_bounds__(512) void init_stats_kernel(
    const unsigned char* __restrict__ mask, float* __restrict__ stats,
    float* __restrict__ cnt)
{
    __shared__ int red;
    if (threadIdx.x == 0) red = 0;
    __syncthreads();
    int c = 0;
    for (int i = threadIdx.x; i < B * L; i += blockDim.x)
        c += mask[i] ? 0 : 1;
    atomicAdd(&red, c);
    __syncthreads();
    if (threadIdx.x < 16) stats[threadIdx.x] = 0.0f;
    if (threadIdx.x == 0) *cnt = (float)red * (float)T;
}

// ---------------------------------------------------------------------------
// Kernel 1: gating network. One block per (b,t) image.
// conv3x3(16->8) + ReLU + mean pool + FC(8->16) + sigmoid -> gate[b,t,16]
// ---------------------------------------------------------------------------
__global__ __launch_bounds__(256) void gating_kernel(
    const float* __restrict__ prev, const float* __restrict__ curr,
    const float* __restrict__ g_w1, const float* __restrict__ g_b1,
    const float* __restrict__ g_w2, const float* __restrict__ g_b2,
    float* __restrict__ gate)
{
    const int bt = blockIdx.x;
    const int b = bt >> 8, t = bt & (T - 1);
    const int tid = threadIdx.x;

    __shared__ float im[C_IN * L];       // 32 KB
    __shared__ float w1[8 * C_IN * 9];   // 4.5 KB
    __shared__ float pooled[8];

    for (int i = tid; i < C_IN * L; i += 256) {
        int ch = i >> 9, l = i & (L - 1);
        const float* src = (ch < NHEAD) ? prev : curr;
        int chh = ch & (NHEAD - 1);
        im[i] = src[(((b * NHEAD + chh) * T) + t) * L + l];
    }
    for (int i = tid; i < 8 * C_IN * 9; i += 256) w1[i] = g_w1[i];
    if (tid < 8) pooled[tid] = 0.0f;
    __syncthreads();

    // 4096 conv outputs, 16 per thread; a thread's 16 stay in one out-channel
    const int oc = tid >> 5;
    float sum = 0.0f;
    for (int s = 0; s < 16; ++s) {
        int pos = ((tid & 31) << 4) + s;
        int hh = pos >> 5, ww = pos & 31;
        float o = g_b1[oc];
        for (int ic = 0; ic < C_IN; ++ic) {
            const float* wrow = &w1[(oc * C_IN + ic) * 9];
            const float* irow = &im[ic * L];
            for (int dy = 0; dy < 3; ++dy) {
                int y = hh + dy - 1;
                if (y < 0 || y >= H) continue;
                for (int dx = 0; dx < 3; ++dx) {
                    int x = ww + dx - 1;
                    if (x < 0 || x >= W) continue;
                    o = fmaf(wrow[dy * 3 + dx], irow[y * W + x], o);
                }
            }
        }
        sum += fmaxf(o, 0.0f);
    }
    atomicAdd(&pooled[oc], sum);   // ds_add_f32
    __syncthreads();

    if (tid < C_IN) {
        float g = g_b2[tid];
        for (int i = 0; i < 8; ++i)
            g = fmaf(pooled[i] * (1.0f / (float)L), g_w2[tid * 8 + i], g);
        gate[(b * T + t) * C_IN + tid] = 1.0f / (1.0f + __expf(-g));
    }
}

// ---------------------------------------------------------------------------
// Kernel 2: exclusive cumsum over t of attns*gate. One block per (b,c).
// ---------------------------------------------------------------------------
__global__ __launch_bounds__(512) void cumsum_kernel(
    const float* __restrict__ prev, const float* __restrict__ curr,
    const float* __restrict__ gate, float* __restrict__ cum)
{
    const int bc = blockIdx.x;
    const int b = bc >> 4, c = bc & (C_IN - 1);
    const float* src = (c < NHEAD) ? prev : curr;
    const int ch = c & (NHEAD - 1);
    const int l = threadIdx.x;

    const float* row0 = src + (size_t)(b * NHEAD + ch) * T * L;
    float* dst0 = cum + (size_t)(b * C_IN + c) * T * L;

    float acc = 0.0f;
    for (int t = 0; t < T; ++t) {
        float g = gate[(b * T + t) * C_IN + c];      // wave-uniform broadcast
        float v = row0[t * L + l] * g;
        dst0[t * L + l] = acc;                        // exclusive
        acc += v;
    }
}

// ---------------------------------------------------------------------------
// Kernel 3: conv5x5(16->32) + bias + ReLU + mask + 1x1 proj(32->8) as an
// implicit GEMM with V_WMMA_F32_16X16X4_F32 (full fp32 matrix path).
// One block (8 waves) per image; each wave computes 8 of 64 16x16 tiles,
// K = 400 in 100 WMMA steps. Also accumulates masked-BN sum / sumsq.
// ---------------------------------------------------------------------------
__global__ __launch_bounds__(256) void conv_proj_kernel(
    const float* __restrict__ cum, const unsigned char* __restrict__ mask,
    const float* __restrict__ conv_w, const float* __restrict__ conv_b,
    const float* __restrict__ proj_w, float* __restrict__ out,
    float* __restrict__ stats)
{
    const int bt = blockIdx.x;
    const int b = bt >> 8, t = bt & (T - 1);
    const int tid = threadIdx.x;

    __shared__ float img[C_IN * 20 * 36];   // zero-padded image, 46080 B
    __shared__ int   koff[KTOT];            // im2col offsets,     1600 B
    __shared__ float ybuf[NHEAD * L];       // projected output,  16384 B
    __shared__ float sstat[16];             // sum / sumsq,          64 B

    for (int i = tid; i < C_IN * 20 * 36; i += 256) img[i] = 0.0f;
    for (int i = tid; i < NHEAD * L; i += 256) ybuf[i] = 0.0f;
    if (tid < 16) sstat[tid] = 0.0f;
    for (int i = tid; i < KTOT; i += 256) {
        int ch = i / 25, r = i % 25, dy = r / 5, dx = r % 5;
        koff[i] = (ch * 20 + dy) * 36 + dx;
    }
    __syncthreads();

    // interior fill of padded image
    for (int i = tid; i < C_IN * L; i += 256) {
        int ch = i >> 9, rem = i & (L - 1);
        int hh = rem >> 5, ww = rem & 31;
        img[(ch * 20 + hh + 2) * 36 + (ww + 2)] =
            cum[(size_t)((b * C_IN + ch) * T + t) * L + rem];
    }
    __syncthreads();

    const int wave = tid >> 5, lane = tid & 31;
    const int kk_half = lane >> 4;          // 0: K={0,1}, 1: K={2,3}
    const int lane15 = lane & 15;
    const v2f* w2 = (const v2f*)conv_w;     // A rows are 8B aligned (k0 even)

    for (int iter = 0; iter < 8; ++iter) {
        const int tile = wave + iter * 8;           // 0..63
        const int m_tile = tile & 1;                // 2 tiles of out channels
        const int n_tile = tile >> 1;               // 32 tiles of positions

        const int mm = m_tile * 16 + lane15;        // A row (out channel)
        const int n  = n_tile * 16 + lane15;        // B col (spatial pos)
        const int base = (n >> 5) * 36 + (n & 31);  // padded-image base

        v8f c = {};
        for (int kb = 0; kb < 100; ++kb) {
            const int k0 = kb * 4 + kk_half * 2;
            v2f a = w2[(mm * KTOT + k0) >> 1];      // global_load_b64, L2-hot
            v2f bf;
            bf.x = img[koff[k0] + base];
            bf.y = img[koff[k0 + 1] + base];
            c = __builtin_amdgcn_wmma_f32_16x16x4_f32(
                    false, a, false, bf, (short)0, c, false, false);
        }

        // epilogue: bias + ReLU + column mask, then 32->8 projection partials
        const int l = n;                            // this lane's column
        const bool masked = mask[b * L + l] != 0;
        float part[NHEAD];
        for (int nh = 0; nh < NHEAD; ++nh) part[nh] = 0.0f;
        for (int r = 0; r < 8; ++r) {
            int m = m_tile * 16 + kk_half * 8 + r;  // C row for this lane
            float v = fmaxf(c[r] + conv_b[m], 0.0f);
            if (masked) v = 0.0f;
            for (int nh = 0; nh < NHEAD; ++nh)
                part[nh] = fmaf(proj_w[nh * DC + m], v, part[nh]);
        }
        for (int nh = 0; nh < NHEAD; ++nh)
            atomicAdd(&ybuf[nh * L + l], part[nh]); // ds_add_f32
    }
    __syncthreads();

    // write pre-BN output + accumulate masked-BN statistics
    for (int i = tid; i < NHEAD * L; i += 256) {
        int nh = i >> 9, l = i & (L - 1);
        float y = ybuf[i];
        out[(size_t)((b * NHEAD + nh) * T + t) * L + l] = y;
        if (!mask[b * L + l]) {
            atomicAdd(&sstat[nh], y);
            atomicAdd(&sstat[8 + nh], y * y);
        }
    }
    __syncthreads();
    if (tid < 16) atomicAdd(&stats[tid], sstat[tid]);  // global_atomic_add_f32
}

// ---------------------------------------------------------------------------
// Kernel 4: apply masked BatchNorm in place on d_out.
// ---------------------------------------------------------------------------
__global__ __launch_bounds__(256) void bn_apply_kernel(
    float* __restrict__ out, const unsigned char* __restrict__ mask,
    const float* __restrict__ stats, const float* __restrict__ cnt_p,
    const float* __restrict__ gamma, const float* __restrict__ beta)
{
    const int i = blockIdx.x * 256 + threadIdx.x;     // 8,388,608 elements
    const int l = i & (L - 1);
    const int nh = (i >> 17) & (NHEAD - 1);
    const int b = i >> 20;
    if (mask[b * L + l]) return;                      // masked stays 0
    const float cnt = *cnt_p;
    const float mean = stats[nh] / cnt;
    float var = stats[8 + nh] / cnt - mean * mean;
    var = fmaxf(var, 0.0f);
    out[i] = (out[i] - mean) * rsqrtf(var + EPS) * gamma[nh] + beta[nh];
}

// ---------------------------------------------------------------------------
extern "C" void kernel_launch(void* const* d_in, const int* in_sizes, int n_in,
                              void* d_out, int out_size, void* d_ws, size_t ws_size,
                              hipStream_t stream) {
    (void)in_sizes; (void)n_in; (void)out_size; (void)ws_size;
    const float*         prev   = (const float*)d_in[0];
    const unsigned char* mask   = (const unsigned char*)d_in[1]; // bool mask
    /* d_in[2] = h (16), unused */
    const float*         curr   = (const float*)d_in[3];
    /* d_in[4] = tgt_vocab, unused */
    const float* conv_w = (const float*)d_in[5];
    const float* conv_b = (const float*)d_in[6];
    const float* proj_w = (const float*)d_in[7];
    const float* g_w1   = (const float*)d_in[8];
    const float* g_b1   = (const float*)d_in[9];
    const float* g_w2   = (const float*)d_in[10];
    const float* g_b2   = (const float*)d_in[11];
    const float* gamma  = (const float*)d_in[12];
    const float* beta   = (const float*)d_in[13];
    float* out = (float*)d_out;
    float* ws  = (float*)d_ws;

    // workspace layout (floats)
    float* gate  = ws;               // 8*256*16 = 32768
    float* stats = ws + 33024;       // 16 (sum[8], sumsq[8])
    float* cnt   = ws + 33056;       // 1
    float* cum   = ws + 33280;       // 8*16*256*512 = 16,777,216 (64 MB)

    init_stats_kernel<<<1, 512, 0, stream>>>(mask, stats, cnt);
    gating_kernel<<<B * T, 256, 0, stream>>>(prev, curr, g_w1, g_b1, g_w2, g_b2, gate);
    cumsum_kernel<<<B * C_IN, 512, 0, stream>>>(prev, curr, gate, cum);
    conv_proj_kernel<<<B * T, 256, 0, stream>>>(cum, mask, conv_w, conv_b, proj_w, out, stats);
    bn_apply_kernel<<<(B * NHEAD * T * L) / 256, 256, 0, stream>>>(out, mask, stats, cnt, gamma, beta);
}